// PowderWorld_46677704573570
// MI455X (gfx1250) — compile-verified
//
#include <hip/hip_runtime.h>

#define NB   8
#define NCH  20
#define HH   512
#define WW   512
#define DENC 14
#define GRVC 15
#define CELLS (NB*HH*WW)     /* 2,097,152 packed bytes per state buffer */
#define WORDS (CELLS/4)      /* 524,288 */

// ---------------- CDNA5 async global->LDS data path (gfx1250) ----------------
// Generic pointers into LDS carry the LDS byte offset in addr[31:0]
// (flat LDS aperture); truncation yields the VDST value
// the async-LDS instructions expect.
__device__ __forceinline__ unsigned lds_off(const void* p) {
  return (unsigned)(unsigned long long)(size_t)p;
}
__device__ __forceinline__ void async_ld_b32(unsigned loff, const void* g) {
  asm volatile("global_load_async_to_lds_b32 %0, %1, off"
               :: "v"(loff), "v"((unsigned long long)(size_t)g) : "memory");
}
__device__ __forceinline__ void async_ld_b128(unsigned loff, const void* g) {
  asm volatile("global_load_async_to_lds_b128 %0, %1, off"
               :: "v"(loff), "v"((unsigned long long)(size_t)g) : "memory");
}
__device__ __forceinline__ void async_wait0() {
  asm volatile("s_wait_asynccnt 0x0" ::: "memory");
}

// ---------------- K1: compress world -> packed bytes, apply stone grav rule --
__global__ __launch_bounds__(256)
void pw_compress(const float* __restrict__ world,
                 unsigned char* __restrict__ dst,
                 unsigned char* __restrict__ didg) {
  int idx = blockIdx.x * 256 + threadIdx.x;   // word index (4 cells)
  int b  = idx >> 16;                         // H*W/4 = 65536 words per batch
  int r  = idx & 65535;
  int h  = r >> 7;                            // W/4 = 128 words per row
  int w0 = (r & 127) << 2;
  const float* wb = world + (size_t)b * NCH * HH * WW;

  int id[4] = {0, 0, 0, 0};
#pragma unroll
  for (int c = 0; c < 14; ++c) {
    float4 v = *(const float4*)(wb + ((size_t)c * HH + h) * WW + w0);
    if (v.x == 1.0f) id[0] = c;
    if (v.y == 1.0f) id[1] = c;
    if (v.z == 1.0f) id[2] = c;
    if (v.w == 1.0f) id[3] = c;
  }
  float4 dv = *(const float4*)(wb + ((size_t)DENC * HH + h) * WW + w0);
  float4 gv = *(const float4*)(wb + ((size_t)GRVC * HH + h) * WW + w0);
  int den[4] = {(int)(dv.x + 0.5f), (int)(dv.y + 0.5f),
                (int)(dv.z + 0.5f), (int)(dv.w + 0.5f)};
  int g[4]   = {gv.x == 1.0f, gv.y == 1.0f, gv.z == 1.0f, gv.w == 1.0f};

  // stone (elem 9): grav = (stone(h-1,w-1)+stone(h-1,w+1) < 2), ZERO padded
  const float* ch9 = wb + (size_t)9 * HH * WW;
#pragma unroll
  for (int j = 0; j < 4; ++j) {
    if (id[j] == 9) {
      int w = w0 + j;
      float s = 0.0f;
      if (h > 0) {
        if (w > 0)       s += ch9[(size_t)(h - 1) * WW + (w - 1)];
        if (w + 1 < WW)  s += ch9[(size_t)(h - 1) * WW + (w + 1)];
      }
      g[j] = (s < 2.0f) ? 1 : 0;
    }
  }
  unsigned out = 0;
#pragma unroll
  for (int j = 0; j < 4; ++j) {
    unsigned byte = (unsigned)(id[j] & 15) | ((unsigned)(den[j] & 7) << 4) |
                    ((unsigned)g[j] << 7);
    out |= byte << (8 * j);
  }
  ((unsigned*)dst)[idx]  = out;
  ((unsigned*)didg)[idx] = 0u;   // init did_gravity
}

// ---------------- K2..K5: one vertical gravity pass (torus wrap) -------------
__global__ __launch_bounds__(128)
void pw_gravity(const unsigned char* __restrict__ src,
                unsigned char* __restrict__ dst,
                unsigned char* __restrict__ didg,
                int curr) {
  __shared__ unsigned srw[3 * (WW / 4)];   // 3 rows of 128 packed words
  int blk = blockIdx.x;
  int b = blk >> 9, h = blk & 511;
  int hm1 = (h + 511) & 511, hp1 = (h + 1) & 511;
  int i = threadIdx.x;                 // word in row, 0..127
  size_t rb = (size_t)b * HH * WW;

  async_ld_b32(lds_off(&srw[0 * 128 + i]), src + rb + (size_t)hm1 * WW + i * 4);
  async_ld_b32(lds_off(&srw[1 * 128 + i]), src + rb + (size_t)h   * WW + i * 4);
  async_ld_b32(lds_off(&srw[2 * 128 + i]), src + rb + (size_t)hp1 * WW + i * 4);
  async_wait0();
  __syncthreads();

  unsigned upw = srw[0 * 128 + i];     // ds_load_b32
  unsigned cuw = srw[1 * 128 + i];
  unsigned dnw = srw[2 * 128 + i];
  unsigned dg = *(const unsigned*)(didg + rb + (size_t)h * WW + i * 4);
  unsigned out = 0;
#pragma unroll
  for (int j = 0; j < 4; ++j) {
    unsigned up = (upw >> (8 * j)) & 255u;
    unsigned cu = (cuw >> (8 * j)) & 255u;
    unsigned dn = (dnw >> (8 * j)) & 255u;
    int den_u = (up >> 4) & 7, den_c = (cu >> 4) & 7, den_d = (dn >> 4) & 7;
    int g_u = up >> 7, g_c = cu >> 7, g_d = dn >> 7;
    bool bel = (den_c == curr) && (den_c > den_d) && g_d && g_c;
    bool abv = (den_u > den_c) && (den_u == curr) && g_u && g_c;
    unsigned nb = bel ? dn : (abv ? up : cu);
    out |= nb << (8 * j);
    if (abv) dg += 1u << (8 * j);
  }
  *(unsigned*)(dst  + rb + (size_t)h * WW + i * 4) = out;
  *(unsigned*)(didg + rb + (size_t)h * WW + i * 4) = dg;
}

// ---------------- K6..K9: one diagonal (sand-like) pass ----------------------
// dx=+1 <=> fall_left=True (ar at (h-1,w+1), bl at (h+1,w-1), match = rand>0.5)
// dx=-1 <=> fall_left=False (mirrored, match = !(rand>0.5))
__global__ __launch_bounds__(128)
void pw_diag(const unsigned char* __restrict__ src,
             unsigned char* __restrict__ dst,
             const unsigned char* __restrict__ didg,
             const float* __restrict__ rnd,
             int elem, int dx) {
  __shared__ unsigned char st [3 * WW];   // state rows: hm1,h,hp1
  __shared__ unsigned char dgr[3 * WW];   // did_gravity rows: hm1,h,hp1
  __shared__ float         rr [2 * WW];   // rand rows: hm1,h

  int blk = blockIdx.x;
  int b = blk >> 9, h = blk & 511;
  int hm1 = (h + 511) & 511, hp1 = (h + 1) & 511;
  int i = threadIdx.x;
  size_t rb = (size_t)b * HH * WW;

  async_ld_b32 (lds_off(&st [0 * WW + i * 4]), src  + rb + (size_t)hm1 * WW + i * 4);
  async_ld_b32 (lds_off(&st [1 * WW + i * 4]), src  + rb + (size_t)h   * WW + i * 4);
  async_ld_b32 (lds_off(&st [2 * WW + i * 4]), src  + rb + (size_t)hp1 * WW + i * 4);
  async_ld_b32 (lds_off(&dgr[0 * WW + i * 4]), didg + rb + (size_t)hm1 * WW + i * 4);
  async_ld_b32 (lds_off(&dgr[1 * WW + i * 4]), didg + rb + (size_t)h   * WW + i * 4);
  async_ld_b32 (lds_off(&dgr[2 * WW + i * 4]), didg + rb + (size_t)hp1 * WW + i * 4);
  async_ld_b128(lds_off(&rr [0 * WW + i * 4]), rnd  + rb + (size_t)hm1 * WW + i * 4);
  async_ld_b128(lds_off(&rr [1 * WW + i * 4]), rnd  + rb + (size_t)h   * WW + i * 4);
  async_wait0();
  __syncthreads();

  unsigned out = 0;
#pragma unroll
  for (int j = 0; j < 4; ++j) {
    int w  = i * 4 + j;
    int wa = (w + dx + WW) & 511;   // ar column (h-1 row)
    int wb = (w - dx + WW) & 511;   // bl column (h+1 row)
    unsigned cu = st[1 * WW + w];   // ds_load_u8 (direct AS3 indexing)
    unsigned ar = st[0 * WW + wa];
    unsigned bl = st[2 * WW + wb];
    bool ndg_c  = dgr[1 * WW + w]  == 0;
    bool ndg_ar = dgr[0 * WW + wa] == 0;
    bool ndg_bl = dgr[2 * WW + wb] == 0;
    float r_c  = rr[1 * WW + w];
    float r_ar = rr[0 * WW + wa];
    bool match_c  = (dx == 1) ? (r_c  > 0.5f) : !(r_c  > 0.5f);
    bool match_ar = (dx == 1) ? (r_ar > 0.5f) : !(r_ar > 0.5f);
    int den_c = (cu >> 4) & 7, den_a = (ar >> 4) & 7, den_b = (bl >> 4) & 7;
    int g_c = cu >> 7, g_a = ar >> 7, g_b = bl >> 7;
    bool blc = ((cu & 15u) == (unsigned)elem) && ndg_bl && ndg_c && match_c &&
               (den_c > den_b) && g_b && g_c;
    bool arc = ((ar & 15u) == (unsigned)elem) && ndg_ar && ndg_c && match_ar &&
               (den_a > den_c) && g_a && g_c;
    unsigned nb = blc ? bl : (arc ? ar : cu);
    out |= nb << (8 * j);
  }
  *(unsigned*)(dst + rb + (size_t)h * WW + i * 4) = out;
}

// ---------------- K10: expand packed bytes -> 20-channel one-hot world -------
__global__ __launch_bounds__(256)
void pw_expand(const unsigned char* __restrict__ src, float* __restrict__ out) {
  int idx = blockIdx.x * 256 + threadIdx.x;   // word index
  int b  = idx >> 16;
  int r  = idx & 65535;
  int h  = r >> 7;
  int w0 = (r & 127) << 2;
  unsigned word = ((const unsigned*)src)[idx];
  int id[4], den[4], g[4];
#pragma unroll
  for (int j = 0; j < 4; ++j) {
    unsigned by = (word >> (8 * j)) & 255u;
    id[j]  = by & 15;
    den[j] = (by >> 4) & 7;
    g[j]   = by >> 7;
  }
  float* ob = out + (size_t)b * NCH * HH * WW;
#pragma unroll
  for (int c = 0; c < NCH; ++c) {
    float4 v;
    if (c < 14) {
      v.x = (id[0] == c) ? 1.0f : 0.0f;  v.y = (id[1] == c) ? 1.0f : 0.0f;
      v.z = (id[2] == c) ? 1.0f : 0.0f;  v.w = (id[3] == c) ? 1.0f : 0.0f;
    } else if (c == DENC) {
      v.x = (float)den[0]; v.y = (float)den[1]; v.z = (float)den[2]; v.w = (float)den[3];
    } else if (c == GRVC) {
      v.x = (float)g[0];   v.y = (float)g[1];   v.z = (float)g[2];   v.w = (float)g[3];
    } else {
      v.x = v.y = v.z = v.w = 0.0f;
    }
    *(float4*)(ob + ((size_t)c * HH + h) * WW + w0) = v;
  }
}

extern "C" void kernel_launch(void* const* d_in, const int* in_sizes, int n_in,
                              void* d_out, int out_size, void* d_ws, size_t ws_size,
                              hipStream_t stream) {
  const float* world = (const float*)d_in[0];
  const float* rnd   = (const float*)d_in[1];
  float* out = (float*)d_out;

  unsigned char* bufA = (unsigned char*)d_ws;
  unsigned char* bufB = bufA + CELLS;
  unsigned char* didg = bufA + 2 * (size_t)CELLS;   // needs 6 MB of d_ws

  pw_compress<<<WORDS / 256, 256, 0, stream>>>(world, bufA, didg);
  pw_gravity <<<NB * HH, 128, 0, stream>>>(bufA, bufB, didg, 0);
  pw_gravity <<<NB * HH, 128, 0, stream>>>(bufB, bufA, didg, 1);
  pw_gravity <<<NB * HH, 128, 0, stream>>>(bufA, bufB, didg, 2);
  pw_gravity <<<NB * HH, 128, 0, stream>>>(bufB, bufA, didg, 3);
  pw_diag    <<<NB * HH, 128, 0, stream>>>(bufA, bufB, didg, rnd,  2, +1);
  pw_diag    <<<NB * HH, 128, 0, stream>>>(bufB, bufA, didg, rnd,  2, -1);
  pw_diag    <<<NB * HH, 128, 0, stream>>>(bufA, bufB, didg, rnd, 12, +1);
  pw_diag    <<<NB * HH, 128, 0, stream>>>(bufB, bufA, didg, rnd, 12, -1);
  pw_expand  <<<WORDS / 256, 256, 0, stream>>>(bufA, out);
}